// GraphConv_37855841747675
// MI455X (gfx1250) — compile-verified
//
#include <hip/hip_runtime.h>
#include <stdint.h>

// ---------------------------------------------------------------------------
// GraphConv on MI455X (gfx1250, wave32):
//   pass1: WMMA fp32 GEMM over gathered neighbors, raw per-query channel max
//          -> d_out, global sum/sumsq -> ws (atomics)
//   pass2: BN affine params (scale/shift) from global stats
//   pass3: in-place  out = leaky_relu(out*scale + shift)
// Valid because BN (gamma=1) + LeakyReLU are monotone increasing per channel,
// so max over neighbors commutes with the post-GEMM pointwise transform.
// ---------------------------------------------------------------------------

typedef float v2f __attribute__((ext_vector_type(2)));
typedef float v8f __attribute__((ext_vector_type(8)));

#define K_NEIGH 32
#define C_FEATN 64
#define C_OUTN  64
#define W_COLS  67      // 3 coords + 64 feats
#define BN_EPSF 1e-5f

__global__ void gc_zero(float* ws) {
  int i = blockIdx.x * blockDim.x + threadIdx.x;
  if (i < 2 * C_OUTN) ws[i] = 0.f;
}

__global__ __launch_bounds__(256)
void gc_main(const float* __restrict__ qpts, const float* __restrict__ spts,
             const float* __restrict__ feat, const float* __restrict__ convw,
             const int*   __restrict__ nbr,  float* __restrict__ outmax,
             float* __restrict__ ws_sum, float* __restrict__ ws_sqs,
             int Q, int NS)
{
  const int lane   = threadIdx.x & 31;
  const int wid    = blockIdx.x * (blockDim.x >> 5) + (threadIdx.x >> 5);
  const int nWaves = gridDim.x * (blockDim.x >> 5);
  const int chHalf = wid & 1;          // which 32 of the 64 output channels
  const int chBase = chHalf * 32;
  const int qw     = wid >> 1;         // query-wave id
  const int nQW    = nWaves >> 1;

  const int lo16 = lane & 15;
  const int hi   = lane >> 4;          // 0: lanes 0-15, 1: lanes 16-31

  // ---- Preload B = W^T in WMMA 4x16 layout, registers, reused all queries.
  // B[c][n] = convw[n*67 + c]. Channel order in y: [coords(3) | feats(64)].
  // k-steps 0..15 cover feat channels c = 3 + 4j .. 3 + 4j+3;
  // k-step 16 covers coord channels c = 0,1,2 zero-padded to 4.
  v2f B0[17], B1[17];
  {
    const float* w0 = convw + (size_t)(chBase + lo16) * W_COLS;
    const float* w1 = w0 + 16 * W_COLS;
    #pragma unroll
    for (int j = 0; j < 16; ++j) {
      int c0 = 3 + 4 * j + 2 * hi;      // rows 4j+2*hi, 4j+2*hi+1 of B
      B0[j].x = w0[c0]; B0[j].y = w0[c0 + 1];
      B1[j].x = w1[c0]; B1[j].y = w1[c0 + 1];
    }
    if (hi == 0) { B0[16].x = w0[0]; B0[16].y = w0[1];
                   B1[16].x = w1[0]; B1[16].y = w1[1]; }
    else         { B0[16].x = w0[2]; B0[16].y = 0.f;
                   B1[16].x = w1[2]; B1[16].y = 0.f; }
  }

  float sum0 = 0.f, sum1 = 0.f, sqs0 = 0.f, sqs1 = 0.f;

  for (int q = qw; q < Q; q += nQW) {
    const float qp0 = qpts[q * 3 + 0];
    const float qp1 = qpts[q * 3 + 1];
    const float qp2 = qpts[q * 3 + 2];
    float mx0 = -__builtin_inff(), mx1 = -__builtin_inff();

    #pragma unroll
    for (int m = 0; m < 2; ++m) {
      // Lane L and L+16 both own neighbor row 16*m + (L&15) of this query.
      const int row  = 16 * m + lo16;
      const int idx  = nbr[(size_t)q * K_NEIGH + row];
      const int pad  = (idx >= NS);          // NS is the zero-row sentinel
      const int sidx = pad ? 0 : idx;
      const float pm = pad ? 0.f : 1.f;

      // ---- Gather A tile (16x68 fp32) directly in WMMA operand layout.
      const float* fr = feat + (size_t)sidx * C_FEATN;
      v2f A[17];
      #pragma unroll
      for (int j = 0; j < 16; ++j) {
        v2f a = *(const v2f*)(fr + 4 * j + 2 * hi);   // 8B-aligned gather
        a.x *= pm; a.y *= pm;
        A[j] = a;
      }
      {
        const float* sp = spts + (size_t)sidx * 3;
        v2f a;
        if (hi == 0) { a.x = (sp[0] - qp0) * pm; a.y = (sp[1] - qp1) * pm; }
        else         { a.x = (sp[2] - qp2) * pm; a.y = 0.f; }
        A[16] = a;
      }

      // ---- Two independent WMMA accumulation chains (N-tiles 0 and 1).
      v8f c0 = {}; v8f c1 = {};
      #pragma unroll
      for (int j = 0; j < 17; ++j) {
        c0 = __builtin_amdgcn_wmma_f32_16x16x4_f32(
                 false, A[j], false, B0[j], (short)0, c0, false, false);
        c1 = __builtin_amdgcn_wmma_f32_16x16x4_f32(
                 false, A[j], false, B1[j], (short)0, c1, false, false);
      }

      // acc[v] at lane L holds z[row = v + 8*hi (+16m)][n = chBase+16t+lo16]
      #pragma unroll
      for (int v = 0; v < 8; ++v) {
        float z0 = c0[v], z1 = c1[v];
        sum0 += z0; sqs0 += z0 * z0;
        sum1 += z1; sqs1 += z1 * z1;
        mx0 = fmaxf(mx0, z0); mx1 = fmaxf(mx1, z1);
      }
    }

    // Fold the two lane-halves (rows v vs v+8) and emit raw channel max.
    mx0 = fmaxf(mx0, __shfl_xor(mx0, 16, 32));
    mx1 = fmaxf(mx1, __shfl_xor(mx1, 16, 32));
    if (hi == 0) {
      outmax[(size_t)q * C_OUTN + chBase + lo16]      = mx0;
      outmax[(size_t)q * C_OUTN + chBase + 16 + lo16] = mx1;
    }
  }

  // ---- Global BN statistics: one atomic per channel per wave.
  sum0 += __shfl_xor(sum0, 16, 32);
  sqs0 += __shfl_xor(sqs0, 16, 32);
  sum1 += __shfl_xor(sum1, 16, 32);
  sqs1 += __shfl_xor(sqs1, 16, 32);
  if (hi == 0) {
    atomicAdd(&ws_sum[chBase + lo16],      sum0);
    atomicAdd(&ws_sqs[chBase + lo16],      sqs0);
    atomicAdd(&ws_sum[chBase + 16 + lo16], sum1);
    atomicAdd(&ws_sqs[chBase + 16 + lo16], sqs1);
  }
}

__global__ void gc_bnparams(const float* __restrict__ ws_sum,
                            const float* __restrict__ ws_sqs,
                            const float* __restrict__ gamma,
                            const float* __restrict__ beta,
                            float* __restrict__ ws_scale,
                            float* __restrict__ ws_shift, float inv_n)
{
  int c = threadIdx.x;
  if (c < C_OUTN) {
    float mean = ws_sum[c] * inv_n;
    float var  = ws_sqs[c] * inv_n - mean * mean;
    float sc   = gamma[c] * rsqrtf(var + BN_EPSF);
    ws_scale[c] = sc;
    ws_shift[c] = beta[c] - mean * sc;
  }
}

__global__ void gc_finalize(float* __restrict__ out,
                            const float* __restrict__ ws_scale,
                            const float* __restrict__ ws_shift, int n)
{
  int i = blockIdx.x * blockDim.x + threadIdx.x;
  if (i < n) {
    int c = i & (C_OUTN - 1);
    float v = out[i] * ws_scale[c] + ws_shift[c];
    out[i] = (v >= 0.f) ? v : 0.1f * v;
  }
}

extern "C" void kernel_launch(void* const* d_in, const int* in_sizes, int n_in,
                              void* d_out, int out_size, void* d_ws, size_t ws_size,
                              hipStream_t stream)
{
  const float* qpts  = (const float*)d_in[0];
  const float* spts  = (const float*)d_in[1];
  const float* feat  = (const float*)d_in[2];
  const float* convw = (const float*)d_in[3];
  const float* gamma = (const float*)d_in[4];
  const float* beta  = (const float*)d_in[5];
  const int*   nbr   = (const int*)d_in[6];
  float* out = (float*)d_out;

  const int Q  = in_sizes[0] / 3;
  const int NS = in_sizes[1] / 3;

  float* ws_sum   = (float*)d_ws;
  float* ws_sqs   = ws_sum + C_OUTN;
  float* ws_scale = ws_sqs + C_OUTN;
  float* ws_shift = ws_scale + C_OUTN;

  gc_zero<<<1, 128, 0, stream>>>(ws_sum);

  const int blocks = 640;      // 5120 waves -> 2560 query-waves, grid-stride
  gc_main<<<blocks, 256, 0, stream>>>(qpts, spts, feat, convw, nbr, out,
                                      ws_sum, ws_sqs, Q, NS);

  const float inv_n = 1.f / ((float)Q * (float)K_NEIGH);
  gc_bnparams<<<1, 64, 0, stream>>>(ws_sum, ws_sqs, gamma, beta,
                                    ws_scale, ws_shift, inv_n);

  const int n = Q * C_OUTN;
  gc_finalize<<<(n + 255) / 256, 256, 0, stream>>>(out, ws_scale, ws_shift, n);
}